// StrongFormPhysicsLoss_29669634081210
// MI455X (gfx1250) — compile-verified
//
#include <hip/hip_runtime.h>
#include <hip/hip_bf16.h>

typedef __attribute__((ext_vector_type(2))) float v2f;
typedef __attribute__((ext_vector_type(8))) float v8f;

#define BLOCK 256

// ---------------------------------------------------------------------------
// Exact 32-lane (wave32) sum using V_WMMA_F32_16X16X4_F32 with a ones B.
// Step 1: A vgpr0 = lane value, vgpr1 = 0  ->  D1[m][n] = v_m + v_{m+16}
// Step 2: 4 chained WMMAs feed pairs of D1 vgprs back as A with C=acc;
//         each adds 4 of the 16 row-sums -> acc = total in every element.
// All adds are f32, multiplies are by 1.0 -> numerically exact wave sum.
// Requires full EXEC: called only from uniform control flow.
// ---------------------------------------------------------------------------
__device__ __forceinline__ float wave_sum_wmma(float v) {
  v2f one; one[0] = 1.0f; one[1] = 1.0f;
  v2f a;   a[0] = v;      a[1] = 0.0f;
  v8f z   = {0.f,0.f,0.f,0.f,0.f,0.f,0.f,0.f};
  v8f d   = __builtin_amdgcn_wmma_f32_16x16x4_f32(false, a, false, one, (short)0, z, false, false);
  v8f acc = {0.f,0.f,0.f,0.f,0.f,0.f,0.f,0.f};
  v2f t;
  t[0] = d[0]; t[1] = d[1];
  acc = __builtin_amdgcn_wmma_f32_16x16x4_f32(false, t, false, one, (short)0, acc, false, false);
  t[0] = d[2]; t[1] = d[3];
  acc = __builtin_amdgcn_wmma_f32_16x16x4_f32(false, t, false, one, (short)0, acc, false, false);
  t[0] = d[4]; t[1] = d[5];
  acc = __builtin_amdgcn_wmma_f32_16x16x4_f32(false, t, false, one, (short)0, acc, false, false);
  t[0] = d[6]; t[1] = d[7];
  acc = __builtin_amdgcn_wmma_f32_16x16x4_f32(false, t, false, one, (short)0, acc, false, false);
  return acc[0];
}

__device__ __forceinline__ float block_sum(float v, float* sbuf) {
  float ws = wave_sum_wmma(v);
  int lane = threadIdx.x & 31;
  int wid  = threadIdx.x >> 5;
  __syncthreads();
  if (lane == 0) sbuf[wid] = ws;
  __syncthreads();
  float t = sbuf[0];
  #pragma unroll
  for (int w = 1; w < BLOCK / 32; ++w) t += sbuf[w];
  return t;
}

__device__ __forceinline__ float wave_max32(float v) {
  #pragma unroll
  for (int off = 16; off > 0; off >>= 1)
    v = fmaxf(v, __shfl_xor(v, off, 32));
  return v;
}

__device__ __forceinline__ float block_max(float v, float* sbuf) {
  float wm = wave_max32(v);
  int lane = threadIdx.x & 31;
  int wid  = threadIdx.x >> 5;
  __syncthreads();
  if (lane == 0) sbuf[wid] = wm;
  __syncthreads();
  float t = sbuf[0];
  #pragma unroll
  for (int w = 1; w < BLOCK / 32; ++w) t = fmaxf(t, sbuf[w]);
  return t;
}

// ---------------------------------------------------------------------------
// Kernel 0: zero the workspace (scatter targets + accumulators).
// ---------------------------------------------------------------------------
__global__ void zero_ws_kernel(float* __restrict__ ws, long long n) {
  long long i = (long long)blockIdx.x * blockDim.x + threadIdx.x;
  long long stride = (long long)gridDim.x * blockDim.x;
  for (; i < n; i += stride) ws[i] = 0.0f;
}

// ---------------------------------------------------------------------------
// Kernel 1: per-element pass.  Streamed element data read non-temporally
// (keep L2 for node gathers + scatter targets); 18 native f32 atomic adds
// per element into L2-resident F_int/M_int/F_ext; per-block reduction of
// kin residual, sum(L), max(|load|), max(L) via WMMA/shuffle epilogue.
// ---------------------------------------------------------------------------
__global__ void elem_kernel(
    const float* __restrict__ phi,
    const float* __restrict__ gux,
    const float* __restrict__ guz,
    const float* __restrict__ gph,
    const float* __restrict__ pE,
    const float* __restrict__ pA,
    const float* __restrict__ pI,
    const float* __restrict__ pL,
    const float* __restrict__ xh,
    const float* __restrict__ ld,
    const int*   __restrict__ conn,
    float* __restrict__ Fint,
    float* __restrict__ Mint,
    float* __restrict__ Fext,
    float* __restrict__ accum,
    int nE)
{
  __shared__ float sbuf[BLOCK / 32];
  float kin = 0.0f, lsum = 0.0f, qmax = 0.0f, lmax = 0.0f;
  long long stride = (long long)gridDim.x * blockDim.x;
  for (long long e = (long long)blockIdx.x * blockDim.x + threadIdx.x; e < nE; e += stride) {
    // prefetch next tile of the streamed element data (global_prefetch_b8)
    __builtin_prefetch(xh   + 3 * (e + stride), 0, 0);
    __builtin_prefetch(ld   + 3 * (e + stride), 0, 0);
    __builtin_prefetch(conn + 2 * (e + stride), 0, 0);
    __builtin_prefetch(pE   + (e + stride),     0, 0);

    int i = __builtin_nontemporal_load(&conn[2 * e + 0]);
    int j = __builtin_nontemporal_load(&conn[2 * e + 1]);

    float x0 = __builtin_nontemporal_load(&xh[3 * e + 0]);
    float x1 = __builtin_nontemporal_load(&xh[3 * e + 1]);
    float x2 = __builtin_nontemporal_load(&xh[3 * e + 2]);

    float E  = __builtin_nontemporal_load(&pE[e]);
    float Aa = __builtin_nontemporal_load(&pA[e]);
    float I2 = __builtin_nontemporal_load(&pI[e]);
    float L  = __builtin_nontemporal_load(&pL[e]);

    float l0 = __builtin_nontemporal_load(&ld[3 * e + 0]);
    float l1 = __builtin_nontemporal_load(&ld[3 * e + 1]);
    float l2 = __builtin_nontemporal_load(&ld[3 * e + 2]);

    // ---- local axes (matches _local_axes) ----
    bool par = fabsf(x1) > 0.99f;
    float r1 = par ? 0.0f : 1.0f;   // ref = par ? e3 : e2, ref = (0, r1, r2)
    float r2 = par ? 1.0f : 0.0f;
    float z0 = x1 * r2 - x2 * r1;
    float z1 = -x0 * r2;
    float z2 = x0 * r1;
    float zn = fmaxf(sqrtf(z0 * z0 + z1 * z1 + z2 * z2), 1e-8f);
    z0 /= zn; z1 /= zn; z2 /= zn;
    float y0 = z1 * x2 - z2 * x1;
    float y1 = z2 * x0 - z0 * x2;
    float y2 = z0 * x1 - z1 * x0;
    float yn = fmaxf(sqrtf(y0 * y0 + y1 * y1 + y2 * y2), 1e-8f);
    y0 /= yn; y1 /= yn; y2 /= yn;

    // ---- gathers (cached: node arrays stay L2-resident) ----
    size_t bi = 3 * (size_t)i, bj = 3 * (size_t)j;
    float gux_i = gux[bi] * x0 + gux[bi + 1] * x1 + gux[bi + 2] * x2;
    float gux_j = gux[bj] * x0 + gux[bj + 1] * x1 + gux[bj + 2] * x2;
    float guz_i = guz[bi] * x0 + guz[bi + 1] * x1 + guz[bi + 2] * x2;
    float guz_j = guz[bj] * x0 + guz[bj + 1] * x1 + guz[bj + 2] * x2;
    float kpi   = gph[bi] * x0 + gph[bi + 1] * x1 + gph[bi + 2] * x2;
    float kpj   = gph[bj] * x0 + gph[bj + 1] * x1 + gph[bj + 2] * x2;
    float phii  = phi[i];
    float phij  = phi[j];

    // ---- element force/moment ----
    float eps_i = x0 * gux_i + x2 * guz_i;
    float eps_j = x0 * gux_j + x2 * guz_j;
    float EA = E * Aa, EI = E * I2;
    float Navg = 0.5f * EA * (eps_i + eps_j);
    float Mi = EI * kpi, Mj = EI * kpj;
    float V  = (Mj - Mi) / L;
    float F0 = Navg * x0 + V * z0;
    float F1 = Navg * x1 + V * z1;
    float F2 = Navg * x2 + V * z2;
    float half_L = 0.5f * L;

    // ---- scatter-add (native global_atomic_add_f32) ----
    unsafeAtomicAdd(&Fint[bi + 0],  F0);
    unsafeAtomicAdd(&Fint[bi + 1],  F1);
    unsafeAtomicAdd(&Fint[bi + 2],  F2);
    unsafeAtomicAdd(&Fint[bj + 0], -F0);
    unsafeAtomicAdd(&Fint[bj + 1], -F1);
    unsafeAtomicAdd(&Fint[bj + 2], -F2);
    unsafeAtomicAdd(&Mint[bi + 0], Mi * y0);
    unsafeAtomicAdd(&Mint[bi + 1], Mi * y1);
    unsafeAtomicAdd(&Mint[bi + 2], Mi * y2);
    unsafeAtomicAdd(&Mint[bj + 0], Mj * y0);
    unsafeAtomicAdd(&Mint[bj + 1], Mj * y1);
    unsafeAtomicAdd(&Mint[bj + 2], Mj * y2);
    unsafeAtomicAdd(&Fext[bi + 0], l0 * half_L);
    unsafeAtomicAdd(&Fext[bi + 1], l1 * half_L);
    unsafeAtomicAdd(&Fext[bi + 2], l2 * half_L);
    unsafeAtomicAdd(&Fext[bj + 0], l0 * half_L);
    unsafeAtomicAdd(&Fext[bj + 1], l1 * half_L);
    unsafeAtomicAdd(&Fext[bj + 2], l2 * half_L);

    // ---- kinematic residual + streamed stats ----
    float dui = z0 * gux_i + z2 * guz_i;
    float duj = z0 * gux_j + z2 * guz_j;
    float ri = phii - dui;
    float rj = phij - duj;
    kin += ri * ri + rj * rj;
    lsum += L;
    lmax  = fmaxf(lmax, L);
    qmax  = fmaxf(qmax, fmaxf(fabsf(l0), fmaxf(fabsf(l1), fabsf(l2))));
  }

  // ---- block epilogue (uniform control flow: WMMA-legal) ----
  float bk = block_sum(kin,  sbuf);
  float bl = block_sum(lsum, sbuf);
  float bq = block_max(qmax, sbuf);
  float bL = block_max(lmax, sbuf);
  if (threadIdx.x == 0) {
    unsafeAtomicAdd(&accum[0], bk);
    unsafeAtomicAdd(&accum[3], bl);
    // non-negative floats: uint bit order == float order
    atomicMax((unsigned int*)&accum[1], __float_as_uint(bq));
    atomicMax((unsigned int*)&accum[2], __float_as_uint(bL));
  }
}

// ---------------------------------------------------------------------------
// Kernel 2: per-node masked reductions.
// ---------------------------------------------------------------------------
__global__ void node_kernel(
    const float* __restrict__ bcD,
    const float* __restrict__ bcR,
    const float* __restrict__ Fint,
    const float* __restrict__ Mint,
    const float* __restrict__ Fext,
    float* __restrict__ accum,
    int nN)
{
  __shared__ float sbuf[BLOCK / 32];
  float sF = 0.f, sFe = 0.f, nfd = 0.f, sM = 0.f, nfr = 0.f, sN = 0.f, npin = 0.f;
  long long stride = (long long)gridDim.x * blockDim.x;
  for (long long n = (long long)blockIdx.x * blockDim.x + threadIdx.x; n < nN; n += stride) {
    size_t b = 3 * (size_t)n;
    float bd = bcD[n], br = bcR[n];
    float fe0 = Fext[b], fe1 = Fext[b + 1], fe2 = Fext[b + 2];
    float f0 = Fint[b] + fe0, f1 = Fint[b + 1] + fe1, f2 = Fint[b + 2] + fe2;
    float m0 = Mint[b], m1 = Mint[b + 1], m2 = Mint[b + 2];
    float fr = f0 * f0 + f1 * f1 + f2 * f2;
    float fe = fe0 * fe0 + fe1 * fe1 + fe2 * fe2;
    float mr = m0 * m0 + m1 * m1 + m2 * m2;
    bool free_d = bd < 0.5f;
    bool free_r = br < 0.5f;
    if (free_d) { sF += fr; sFe += fe; nfd += 1.0f; }
    if (free_r) { sM += mr; nfr += 1.0f; }
    if (!free_d && free_r) { sN += mr; npin += 1.0f; }
  }
  float v;
  v = block_sum(sF,   sbuf); if (threadIdx.x == 0) unsafeAtomicAdd(&accum[4],  v);
  v = block_sum(sFe,  sbuf); if (threadIdx.x == 0) unsafeAtomicAdd(&accum[5],  v);
  v = block_sum(nfd,  sbuf); if (threadIdx.x == 0) unsafeAtomicAdd(&accum[6],  v);
  v = block_sum(sM,   sbuf); if (threadIdx.x == 0) unsafeAtomicAdd(&accum[7],  v);
  v = block_sum(nfr,  sbuf); if (threadIdx.x == 0) unsafeAtomicAdd(&accum[8],  v);
  v = block_sum(sN,   sbuf); if (threadIdx.x == 0) unsafeAtomicAdd(&accum[9],  v);
  v = block_sum(npin, sbuf); if (threadIdx.x == 0) unsafeAtomicAdd(&accum[10], v);
}

// ---------------------------------------------------------------------------
// Kernel 3: scalar finalize.
// ---------------------------------------------------------------------------
__global__ void finalize_kernel(const float* __restrict__ accum,
                                float* __restrict__ out, float nE) {
  if (blockIdx.x == 0 && threadIdx.x == 0) {
    float kin  = accum[0];
    float qmax = accum[1];   // bit-pattern max of non-negative floats
    float lmax = accum[2];
    float lsum = accum[3];
    float sF   = accum[4];
    float sFe  = accum[5];
    float nfd  = fmaxf(accum[6], 1.0f);
    float sM   = accum[7];
    float nfr  = fmaxf(accum[8], 1.0f);
    float sN   = accum[9];
    float npin = accum[10];

    float Fchar = fmaxf(sqrtf(sFe / (3.0f * nfd)), 1.0f);
    float qm    = fmaxf(qmax, 1.0f);
    float Mchar = fmaxf(qm * lmax * lsum * 0.125f, 1.0f);

    float Lforce = sF / (3.0f * nfd * Fchar * Fchar);
    float Lmom   = sM / (3.0f * nfr * Mchar * Mchar);
    float Lneu   = (npin > 0.0f) ? sN / (3.0f * fmaxf(npin, 1.0f) * Mchar * Mchar) : 0.0f;
    float Lkin   = kin / (2.0f * nE);

    out[0] = 1.0f * Lforce + 1.0f * Lmom + 1.0f * Lneu + 0.1f * Lkin;
  }
}

// ---------------------------------------------------------------------------
extern "C" void kernel_launch(void* const* d_in, const int* in_sizes, int n_in,
                              void* d_out, int out_size, void* d_ws, size_t ws_size,
                              hipStream_t stream) {
  const float* phi  = (const float*)d_in[0];
  const float* gux  = (const float*)d_in[1];
  const float* guz  = (const float*)d_in[2];
  const float* gph  = (const float*)d_in[3];
  const float* pE   = (const float*)d_in[4];
  const float* pA   = (const float*)d_in[5];
  const float* pI   = (const float*)d_in[6];
  const float* pL   = (const float*)d_in[7];
  const float* xh   = (const float*)d_in[8];
  const float* ld   = (const float*)d_in[9];
  const float* bcD  = (const float*)d_in[10];
  const float* bcR  = (const float*)d_in[11];
  const int*   conn = (const int*)d_in[12];

  int nN = in_sizes[0];   // phi is (N,1)
  int nE = in_sizes[4];   // prop_E is (E,)

  float* ws    = (float*)d_ws;
  float* Fint  = ws;
  float* Mint  = ws + 3 * (size_t)nN;
  float* Fext  = ws + 6 * (size_t)nN;
  float* accum = ws + 9 * (size_t)nN;
  long long ztotal = 9LL * nN + 16;

  zero_ws_kernel<<<4096, BLOCK, 0, stream>>>(ws, ztotal);
  elem_kernel<<<3072, BLOCK, 0, stream>>>(phi, gux, guz, gph, pE, pA, pI, pL,
                                          xh, ld, conn, Fint, Mint, Fext, accum, nE);
  node_kernel<<<2048, BLOCK, 0, stream>>>(bcD, bcR, Fint, Mint, Fext, accum, nN);
  finalize_kernel<<<1, 32, 0, stream>>>(accum, (float*)d_out, (float)nE);
}